// GCN_20263655703368
// MI455X (gfx1250) — compile-verified
//
#include <hip/hip_runtime.h>

typedef float v2f __attribute__((ext_vector_type(2)));
typedef float v8f __attribute__((ext_vector_type(8)));

#define F_IN  256
#define F_HID 96
#define F_OUT 40

// ---------------- degree / normalization ----------------

__global__ void k_deg_init(float* __restrict__ deg, int n) {
  int i = blockIdx.x * blockDim.x + threadIdx.x;
  if (i < n) deg[i] = 1.0f;  // self-loop contribution
}

__global__ void k_deg_count(const long long* __restrict__ dst, float* __restrict__ deg, int E) {
  int e = blockIdx.x * blockDim.x + threadIdx.x;
  if (e < E) {
    int d = (int)dst[e];
    __hip_atomic_fetch_add(&deg[d], 1.0f, __ATOMIC_RELAXED, __HIP_MEMORY_SCOPE_AGENT);
  }
}

__global__ void k_rsqrt_inplace(float* __restrict__ deg, int n) {
  int i = blockIdx.x * blockDim.x + threadIdx.x;
  if (i < n) deg[i] = rsqrtf(deg[i]);   // deg >= 1 always
}

__global__ void k_zero(float* __restrict__ p, int n) {
  int i = blockIdx.x * blockDim.x + threadIdx.x;
  if (i < n) p[i] = 0.0f;
}

// ---------------- dense GEMM via V_WMMA_F32_16X16X4_F32 ----------------
// C[M, ncols] = A[M, K] @ B[K, ncols], row-major.
// Block = WAVES waves; each wave owns one 16x16 output tile (same N-tile,
// consecutive M-tiles). B tile [K x 16] staged in LDS once per block and
// shared by all waves. Masked N-tile columns are clamped (loads always
// valid) and only the final store is predicated -> no exec-mask churn in
// the inner loop.
template <int K, int WAVES>
__global__ __launch_bounds__(WAVES * 32)
void k_gemm_wmma_f32(const float* __restrict__ A, const float* __restrict__ B,
                     float* __restrict__ C, int ncols) {
  __shared__ float Bs[K * 16];

  const int tid  = threadIdx.x;
  const int lane = tid & 31;
  const int wave = tid >> 5;
  const int lo   = lane & 15;
  const int kh   = lane >> 4;                  // 0: K {k,k+1}; 1: K {k+2,k+3}
  const int n0   = blockIdx.y * 16;

  // cooperative stage of B[:, n0:n0+16] with column clamp
  for (int i = tid; i < K * 16; i += WAVES * 32) {
    int kk = i >> 4, j = i & 15;
    int nc = n0 + j;
    if (nc >= ncols) nc = ncols - 1;           // clamp: garbage cols masked at store
    Bs[i] = B[(long)kk * ncols + nc];
  }
  __syncthreads();

  const int mtile = blockIdx.x * WAVES + wave;
  const float* __restrict__ arow = A + (long)(mtile * 16 + lo) * K;

  v8f acc = {};
#pragma unroll 4
  for (int k = 0; k < K; k += 4) {
    const int ka = k + 2 * kh;
    v2f a = *(const v2f*)(arow + ka);          // A[M=lo][ka], [ka+1]  (8B aligned)
    v2f b;
    b.x = Bs[ka * 16 + lo];                    // B[ka][n]
    b.y = Bs[ka * 16 + 16 + lo];               // B[ka+1][n]
    acc = __builtin_amdgcn_wmma_f32_16x16x4_f32(
        /*neg_a=*/false, a, /*neg_b=*/false, b,
        /*c_mod=*/(short)0, acc, /*reuse_a=*/false, /*reuse_b=*/false);
  }

  const int n = n0 + lo;
  if (n < ncols) {
    const int mb = mtile * 16 + kh * 8;        // D: VGPR r -> row r + 8*kh, col lo
#pragma unroll
    for (int r = 0; r < 8; ++r)
      C[(long)(mb + r) * ncols + n] = acc[r];
  }
}

// ---------------- edge scatter: agg[dst] += h[src] * dinv[src]*dinv[dst] ----------------
// blockDim.x == F (feature count), one block per edge.
__global__ void k_scatter(const long long* __restrict__ ei, const float* __restrict__ dinv,
                          const float* __restrict__ h, float* __restrict__ agg, int E, int F) {
  const int e = blockIdx.x;
  const int f = threadIdx.x;
  const int s = (int)ei[e];
  const int d = (int)ei[(long)E + e];
  const float w = dinv[s] * dinv[d];
  const float v = h[(long)s * F + f] * w;
  __hip_atomic_fetch_add(&agg[(long)d * F + f], v, __ATOMIC_RELAXED, __HIP_MEMORY_SCOPE_AGENT);
}

// h[i,f] = relu(agg[i,f] + h[i,f]*dinv[i]^2 + b[f])   (self-loop + bias + relu, in place)
__global__ void k_bias_relu_self(float* __restrict__ h, const float* __restrict__ agg,
                                 const float* __restrict__ dinv, const float* __restrict__ b,
                                 int total, int F) {
  int idx = blockIdx.x * blockDim.x + threadIdx.x;
  if (idx >= total) return;
  int r = idx / F, f = idx - r * F;
  float di = dinv[r];
  float v = agg[idx] + h[idx] * di * di + b[f];
  h[idx] = v > 0.0f ? v : 0.0f;
}

// out[i,f] += h3[i,f]*dinv[i]^2 + b[f]   (layer-2 self-loop + bias)
__global__ void k_final(float* __restrict__ out, const float* __restrict__ h3,
                        const float* __restrict__ dinv, const float* __restrict__ b,
                        int total, int F) {
  int idx = blockIdx.x * blockDim.x + threadIdx.x;
  if (idx >= total) return;
  int r = idx / F, f = idx - r * F;
  float di = dinv[r];
  out[idx] += h3[idx] * di * di + b[f];
}

extern "C" void kernel_launch(void* const* d_in, const int* in_sizes, int n_in,
                              void* d_out, int out_size, void* d_ws, size_t ws_size,
                              hipStream_t stream) {
  const float*     x  = (const float*)d_in[0];
  const long long* ei = (const long long*)d_in[1];     // [2, E] int64
  const float*     W1 = (const float*)d_in[2];
  const float*     b1 = (const float*)d_in[3];
  const float*     W2 = (const float*)d_in[4];
  const float*     b2 = (const float*)d_in[5];
  float*           out = (float*)d_out;

  const int n = in_sizes[0] / F_IN;   // 50000 (= 16 * 3125, 3125 = 5^5)
  const int E = in_sizes[1] / 2;      // 800000

  // workspace layout (floats): dinv | h1(=h2) | agg1(=h3)
  float* ws   = (float*)d_ws;
  const long nAl  = ((long)n + 63) & ~63L;
  float* dinv = ws;
  float* h1   = ws + nAl;
  float* agg1 = h1 + (long)n * F_HID;

  const int T = 256;
  const int WV = 5;                    // waves per GEMM block; 3125 M-tiles / 5 = 625
  const int mblk = (n / 16) / WV;      // 625

  // 1) degree + dinv
  k_deg_init <<<(n + T - 1) / T, T, 0, stream>>>(dinv, n);
  k_deg_count<<<(E + T - 1) / T, T, 0, stream>>>(ei + E, dinv, E);
  k_rsqrt_inplace<<<(n + T - 1) / T, T, 0, stream>>>(dinv, n);

  // 2) layer 1: h1 = x @ W1 (WMMA), scatter, bias+relu+self-loop
  k_zero<<<((n * F_HID) + T - 1) / T, T, 0, stream>>>(agg1, n * F_HID);
  {
    dim3 g(mblk, (F_HID + 15) / 16);   // 625 x 6
    k_gemm_wmma_f32<F_IN, WV><<<g, WV * 32, 0, stream>>>(x, W1, h1, F_HID);
  }
  k_scatter<<<E, F_HID, 0, stream>>>(ei, dinv, h1, agg1, E, F_HID);
  k_bias_relu_self<<<((n * F_HID) + T - 1) / T, T, 0, stream>>>(h1, agg1, dinv, b1,
                                                                n * F_HID, F_HID);

  // 3) layer 2: h3 = h2 @ W2 (WMMA, into agg1 buffer), scatter into out, finish
  {
    dim3 g(mblk, (F_OUT + 15) / 16);   // 625 x 3 (last tile masked)
    k_gemm_wmma_f32<F_HID, WV><<<g, WV * 32, 0, stream>>>(h1, W2, agg1, F_OUT);
  }
  k_zero<<<((n * F_OUT) + T - 1) / T, T, 0, stream>>>(out, n * F_OUT);
  k_scatter<<<E, F_OUT, 0, stream>>>(ei, dinv, agg1, out, E, F_OUT);
  k_final<<<((n * F_OUT) + T - 1) / T, T, 0, stream>>>(out, agg1, dinv, b2,
                                                       n * F_OUT, F_OUT);
}